// PCAM_Model_2_levels_w_Att_10565619549016
// MI455X (gfx1250) — compile-verified
//
#include <hip/hip_runtime.h>
#include <hip/hip_bf16.h>

typedef __attribute__((ext_vector_type(2))) float v2f;
typedef __attribute__((ext_vector_type(8))) float v8f;

#define D_K   512   // embedding dim
#define ATT   128   // attention units
#define NBAGS 256
#define NINST 131072
#define KSTEPS (D_K / 4)          // 128 WMMA K-steps
#define WPACK_ELEMS (KSTEPS * 8 * 64)   // 65536 floats = 256 KB per weight

// ---------------------------------------------------------------------------
// Kernel 0: zero scratch accumulators
// ---------------------------------------------------------------------------
__global__ void zero_kernel(float* __restrict__ p, int n) {
    int i = blockIdx.x * blockDim.x + threadIdx.x;
    if (i < n) p[i] = 0.0f;
}

// ---------------------------------------------------------------------------
// Pack W [512,128] into per-lane WMMA B-fragment order so the hot loop can
// fetch each fragment with one contiguous global_load_b64:
//   Wp[(kstep*8 + n)*64 + lane*2 + {0,1}] =
//       W[(4*kstep + 2*(lane>>4) + {0,1})*128 + n*16 + (lane&15)]
// ---------------------------------------------------------------------------
__global__ void pack_w_kernel(const float* __restrict__ W, float* __restrict__ Wp) {
    int idx = blockIdx.x * blockDim.x + threadIdx.x;   // 0 .. 128*8*32-1
    if (idx >= KSTEPS * 8 * 32) return;
    int lane  = idx & 31;
    int n     = (idx >> 5) & 7;
    int kstep = idx >> 8;
    int col = lane & 15;
    int kb  = (lane >> 4) * 2;
    int k   = kstep * 4;
    Wp[(size_t)(kstep * 8 + n) * 64 + lane * 2 + 0] = W[(size_t)(k + kb)     * ATT + n * 16 + col];
    Wp[(size_t)(kstep * 8 + n) * 64 + lane * 2 + 1] = W[(size_t)(k + kb + 1) * ATT + n * 16 + col];
}

// ---------------------------------------------------------------------------
// Kernel 1/3: fused attention head.
//   logits = tanh(emb @ W + bA) @ wO + bO        (per row)
//   att    = sigmoid(logit)
//   eOut[r] = exp(att);  atomicAdd(sOut, sum of exp)   [softmax denominator]
// One wave per 32 rows: two 16-row tiles x eight 16-col tiles over K=512 via
// V_WMMA_F32_16X16X4_F32 (fp32 in, fp32 accum). B comes pre-packed (Wp) so
// each fragment is one b64 load: 10 VMEM per 16 WMMAs in the steady state.
// ---------------------------------------------------------------------------
__global__ __launch_bounds__(32)
void att_head_kernel(const float* __restrict__ emb,   // [nRows, 512]
                     const float* __restrict__ Wp,    // packed [128*8*64]
                     const float* __restrict__ bA,    // [128]
                     const float* __restrict__ wO,    // [128]
                     const float* __restrict__ bO,    // [1]
                     float* __restrict__ eOut,        // [nRows]
                     float* __restrict__ sOut)        // [1]
{
    const int lane = threadIdx.x;     // 0..31
    const int col  = lane & 15;       // A: M index   B: N index
    const int hi   = lane >> 4;       // half-wave
    const int kb   = hi * 2;          // K sub-offset within a K=4 step
    const int r0   = blockIdx.x * 32; // two 16-row tiles

    v8f acc[2][8];
    #pragma unroll
    for (int t = 0; t < 2; ++t)
        #pragma unroll
        for (int n = 0; n < 8; ++n) { v8f z = {}; acc[t][n] = z; }

    // A-fragment base pointers: row = r0 + col (+16), k offset = kb
    const float* aRow0 = emb + (size_t)(r0 + col) * D_K + kb;
    const float* aRow1 = aRow0 + (size_t)16 * D_K;
    const float* wpl   = Wp + lane * 2;   // per-lane packed-B base

    for (int ks = 0; ks < KSTEPS; ++ks) {
        v2f a0 = *(const v2f*)(aRow0 + ks * 4);
        v2f a1 = *(const v2f*)(aRow1 + ks * 4);
        const float* bp = wpl + (size_t)ks * (8 * 64);
        #pragma unroll
        for (int n = 0; n < 8; ++n) {
            v2f b = *(const v2f*)(bp + n * 64);
            acc[0][n] = __builtin_amdgcn_wmma_f32_16x16x4_f32(
                false, a0, false, b, (short)0, acc[0][n], false, false);
            acc[1][n] = __builtin_amdgcn_wmma_f32_16x16x4_f32(
                false, a1, false, b, (short)0, acc[1][n], false, false);
        }
    }

    // Epilogue: per row, logit = sum_j tanh(acc + bA[j]) * wO[j]
    // C/D layout: vgpr v, lanes 0-15 -> M=v, lanes 16-31 -> M=v+8; N = col.
    const float bo = bO[0];
    #pragma unroll
    for (int t = 0; t < 2; ++t) {
        float part[8];
        #pragma unroll
        for (int v = 0; v < 8; ++v) part[v] = 0.0f;
        #pragma unroll
        for (int n = 0; n < 8; ++n) {
            float bias = bA[n * 16 + col];
            float w    = wO[n * 16 + col];
            #pragma unroll
            for (int v = 0; v < 8; ++v)
                part[v] += tanhf(acc[t][n][v] + bias) * w;
        }
        // butterfly reduce over the 16 lanes of each half-wave (wave32)
        #pragma unroll
        for (int v = 0; v < 8; ++v) {
            #pragma unroll
            for (int m = 1; m < 16; m <<= 1)
                part[v] += __shfl_xor(part[v], m, 16);
        }
        if (col == 0) {   // lane 0 -> rows M=0..7, lane 16 -> rows M=8..15
            float s = 0.0f;
            #pragma unroll
            for (int v = 0; v < 8; ++v) {
                float logit = part[v] + bo;
                float att   = 1.0f / (1.0f + expf(-logit)); // sigmoid
                float e     = expf(att);                    // softmax numerator
                int row = r0 + t * 16 + hi * 8 + v;
                eOut[row] = e;
                s += e;
            }
            atomicAdd(sOut, s);
        }
    }
}

// ---------------------------------------------------------------------------
// Kernel 2: per-bag attention-weighted segment sum (labels sorted/contiguous)
//   bagEmb[b, :] = sum_{lab[i]==b} (e1[i]/S) * emb[i, :]
// 512 threads = one column each; 64 rows per block; local accumulate, flush
// with atomicAdd only on bag change (few per chunk since labels are sorted).
// ---------------------------------------------------------------------------
__global__ __launch_bounds__(512)
void seg_sum_kernel(const float* __restrict__ emb,   // [N, 512]
                    const int*   __restrict__ lab,   // [N]
                    const float* __restrict__ e1,    // [N]
                    const float* __restrict__ S1,    // [1]
                    float* __restrict__ bagEmb,      // [256, 512] (pre-zeroed)
                    int rowsPerBlock, int nRows)
{
    const int c = threadIdx.x;
    const float invS = 1.0f / S1[0];
    int r0 = blockIdx.x * rowsPerBlock;
    int r1 = r0 + rowsPerBlock;
    if (r1 > nRows) r1 = nRows;

    int   cur = -1;
    float acc = 0.0f;
    for (int r = r0; r < r1; ++r) {
        int   b = lab[r];
        float w = e1[r] * invS;
        if (b != cur) {
            if (cur >= 0) atomicAdd(&bagEmb[(size_t)cur * D_K + c], acc);
            cur = b; acc = 0.0f;
        }
        acc += w * emb[(size_t)r * D_K + c];
    }
    if (cur >= 0) atomicAdd(&bagEmb[(size_t)cur * D_K + c], acc);
}

// ---------------------------------------------------------------------------
// Kernel 4: bag-level softmax combine + linear classifier + sigmoid
//   outer = (e2/S2) @ bagEmb    [512]
//   h     = outer @ W_c + b_c   [512]
//   pred  = sigmoid(h . w_out + b_out)
// ---------------------------------------------------------------------------
__global__ __launch_bounds__(512)
void final_kernel(const float* __restrict__ bagEmb,  // [256, 512]
                  const float* __restrict__ e2,      // [256]
                  const float* __restrict__ S2,      // [1]
                  const float* __restrict__ Wc,      // [512, 512]
                  const float* __restrict__ bc,      // [512]
                  const float* __restrict__ wOut,    // [512]
                  const float* __restrict__ bOut,    // [1]
                  float* __restrict__ out)           // [1]
{
    __shared__ float sOuter[512];
    __shared__ float sRed[512];
    const int t = threadIdx.x;
    const float invS = 1.0f / S2[0];

    float o = 0.0f;
    for (int b = 0; b < NBAGS; ++b)
        o += (e2[b] * invS) * bagEmb[(size_t)b * D_K + t];
    sOuter[t] = o;
    __syncthreads();

    float h = bc[t];
    for (int d = 0; d < D_K; ++d)
        h += sOuter[d] * Wc[(size_t)d * 512 + t];
    sRed[t] = h * wOut[t];
    __syncthreads();

    for (int s = 256; s > 0; s >>= 1) {
        if (t < s) sRed[t] += sRed[t + s];
        __syncthreads();
    }
    if (t == 0) out[0] = 1.0f / (1.0f + expf(-(sRed[0] + bOut[0])));
}

// ---------------------------------------------------------------------------
// Workspace layout (floats):
//   [0,       131072)   e1   = exp(att1) per instance
//   [131072,  262144)   bagEmb [256,512]        (zeroed each call)
//   [262144,  262400)   e2   = exp(att2) per bag
//   262400              S1
//   262401              S2
//   [262408,  327944)   Wp1  packed W_a1 fragments (256 KB)
//   [327944,  393480)   Wp2  packed W_a2 fragments (256 KB)
// ---------------------------------------------------------------------------
extern "C" void kernel_launch(void* const* d_in, const int* in_sizes, int n_in,
                              void* d_out, int out_size, void* d_ws, size_t ws_size,
                              hipStream_t stream) {
    const float* first_emb = (const float*)d_in[0];
    const int*   first_lab = (const int*)  d_in[1];
    const float* W_a1 = (const float*)d_in[2];
    const float* b_a1 = (const float*)d_in[3];
    const float* w_o1 = (const float*)d_in[4];
    const float* b_o1 = (const float*)d_in[5];
    const float* W_a2 = (const float*)d_in[6];
    const float* b_a2 = (const float*)d_in[7];
    const float* w_o2 = (const float*)d_in[8];
    const float* b_o2 = (const float*)d_in[9];
    const float* W_c   = (const float*)d_in[10];
    const float* b_c   = (const float*)d_in[11];
    const float* w_out = (const float*)d_in[12];
    const float* b_out = (const float*)d_in[13];
    float* out = (float*)d_out;

    float* wsf    = (float*)d_ws;
    float* e1     = wsf;
    float* bagEmb = wsf + NINST;
    float* e2     = wsf + NINST + NBAGS * D_K;
    float* S1     = wsf + NINST + NBAGS * D_K + NBAGS;
    float* S2     = S1 + 1;
    float* Wp1    = wsf + 262408;
    float* Wp2    = Wp1 + WPACK_ELEMS;

    // zero bagEmb + e2 + S1 + S2 (contiguous region after e1)
    int nZero = NBAGS * D_K + NBAGS + 2;
    zero_kernel<<<(nZero + 255) / 256, 256, 0, stream>>>(bagEmb, nZero);

    // pack both attention weight matrices into WMMA fragment order
    pack_w_kernel<<<(KSTEPS * 8 * 32 + 255) / 256, 256, 0, stream>>>(W_a1, Wp1);
    pack_w_kernel<<<(KSTEPS * 8 * 32 + 255) / 256, 256, 0, stream>>>(W_a2, Wp2);

    // level-1 attention head (WMMA) over all instances
    att_head_kernel<<<NINST / 32, 32, 0, stream>>>(
        first_emb, Wp1, b_a1, w_o1, b_o1, e1, S1);

    // attention-weighted segment sum -> bag embeddings
    seg_sum_kernel<<<NINST / 64, 512, 0, stream>>>(
        first_emb, first_lab, e1, S1, bagEmb, 64, NINST);

    // level-2 attention head (same WMMA kernel) over bags
    att_head_kernel<<<NBAGS / 32, 32, 0, stream>>>(
        bagEmb, Wp2, b_a2, w_o2, b_o2, e2, S2);

    // bag softmax combine + classifier
    final_kernel<<<1, 512, 0, stream>>>(
        bagEmb, e2, S2, W_c, b_c, w_out, b_out, out);
}